// Trans_infer_rnn_tree_51539607552487
// MI455X (gfx1250) — compile-verified
//
#include <hip/hip_runtime.h>
#include <math.h>

typedef __attribute__((ext_vector_type(2))) float v2f;
typedef __attribute__((ext_vector_type(8))) float v8f;

#define CH 256
#define HID 10
#define BATCH 8
#define HH 128
#define WWID 128
#define HWSZ (HH*WWID)
#define NPIX (BATCH*HWSZ)
#define TILE 16
#define NTILES (NPIX/TILE)          // 8192
#define TILES_PER_WG 8
#define NWG (NTILES/TILES_PER_WG)   // 1024
#define NWAVES 20
#define NTHREADS (NWAVES*32)        // 640
#define FSTR 260                    // padded channel stride for LDS feature tiles
#define YSTR 17                     // padded pixel stride for LDS Y
#define NROWS 320                   // 192 (leaf, padded) + 80 (up/low) + 48 (full, padded)
#define ASTR 132                    // padded col stride for LDS A (conflict-free: 132%64==4)
#define KLDS 128                    // K columns held in LDS (s = 0..31)

__device__ __forceinline__ float sigmoidf(float x) { return 1.0f / (1.0f + expf(-x)); }

__device__ __forceinline__ const float* wrow_ptr(int g,
    const float* W_leaf, const float* W_up, const float* Wf_up,
    const float* W_low, const float* Wf_low, const float* W_full, const float* Wf_full,
    float& scale)
{
    // Global row map: [0,180) W_leaf | [180,192) pad | [192,222) W_up | [222,232) Wf_up
    //                 [232,262) W_low | [262,272) Wf_low | [272,302) W_full | [302,312) Wf_full | pad
    scale = 1.0f;
    if (g < 180)  { return W_leaf + (size_t)g*257; }
    if (g < 192)  { scale = 0.0f; return W_leaf; }
    if (g < 222)  { return W_up   + (size_t)(g-192)*267; }
    if (g < 232)  { return Wf_up  + (size_t)(g-222)*267; }
    if (g < 262)  { return W_low  + (size_t)(g-232)*267; }
    if (g < 272)  { return Wf_low + (size_t)(g-262)*267; }
    if (g < 302)  { return W_full + (size_t)(g-272)*267; }
    if (g < 312)  { return Wf_full+ (size_t)(g-302)*267; }
    scale = 0.0f; return W_leaf;
}

__global__ __launch_bounds__(NTHREADS, 1)
void tree_convlstm_fused(
    const float* __restrict__ seg_p, const float* __restrict__ seg_h, const float* __restrict__ seg_f,
    const float* __restrict__ p_fea, const float* __restrict__ h_fea, const float* __restrict__ f_fea,
    const float* __restrict__ W_leaf, const float* __restrict__ W_up,  const float* __restrict__ Wf_up,
    const float* __restrict__ W_low, const float* __restrict__ Wf_low,
    const float* __restrict__ W_full, const float* __restrict__ Wf_full,
    float* __restrict__ out)
{
    __shared__ float AwLds[NROWS * ASTR];     // packed A operand, K columns 0..127
    __shared__ float feat[3 * TILE * FSTR];   // [feature][pixel][channel]
    __shared__ float Y[NROWS * YSTR];         // GEMM results per pixel-tile
    __shared__ float leafH[6 * HID * TILE];
    __shared__ float leafC[6 * HID * TILE];
    __shared__ float lowHs[HID * TILE];
    __shared__ float upCs[HID * TILE];
    __shared__ float lowCs[HID * TILE];
    __shared__ float smallw[1500];            // seg columns + 10-dim h-weight tails

    const int tid    = threadIdx.x;
    const int wave   = tid >> 5;
    const int lane   = tid & 31;
    const int pcol   = lane & 15;             // pixel within tile / M-row within tile
    const int khalf  = lane >> 4;             // 0: K+0..1, 1: K+2..3
    const int kdelta = khalf * 2;

    // ---- small weights (columns 256..266 of each matrix) ----
    for (int idx = tid; idx < 1500; idx += NTHREADS) {
        float v;
        if (idx < 180)       { v = W_leaf[idx*257 + 256]; }
        else if (idx < 510)  { int q = idx-180;  v = W_up  [(q/11)*267 + 256 + (q%11)]; }
        else if (idx < 620)  { int q = idx-510;  v = Wf_up [(q/11)*267 + 256 + (q%11)]; }
        else if (idx < 950)  { int q = idx-620;  v = W_low [(q/11)*267 + 256 + (q%11)]; }
        else if (idx < 1060) { int q = idx-950;  v = Wf_low[(q/11)*267 + 256 + (q%11)]; }
        else if (idx < 1390) { int q = idx-1060; v = W_full[(q/11)*267 + 256 + (q%11)]; }
        else                 { int q = idx-1390; v = Wf_full[(q/11)*267 + 256 + (q%11)]; }
        smallw[idx] = v;
    }

    // ---- pack A for K columns 0..127 into LDS (cooperative, once per WG) ----
    for (int idx = tid; idx < NROWS * KLDS; idx += NTHREADS) {
        int g = idx >> 7;          // row
        int c = idx & (KLDS - 1);  // col 0..127
        float scale;
        const float* rp = wrow_ptr(g, W_leaf, W_up, Wf_up, W_low, Wf_low, W_full, Wf_full, scale);
        AwLds[g*ASTR + c] = rp[c] * scale;
    }

    // ---- per-wave A operand for K columns 128..255, register-resident (64 VGPRs) ----
    const int m0   = wave * 16;
    const int grow = m0 + pcol;
    float ascale;
    const float* rowptr = wrow_ptr(grow, W_leaf, W_up, Wf_up, W_low, Wf_low, W_full, Wf_full, ascale);

    v2f Ar[32];
#pragma unroll
    for (int s = 0; s < 32; ++s) {
        Ar[s].x = rowptr[KLDS + 4*s + kdelta]     * ascale;
        Ar[s].y = rowptr[KLDS + 4*s + kdelta + 1] * ascale;
    }

    const int fsel = (wave < 12) ? 0 : ((wave < 17) ? 1 : 2);  // p_fea / h_fea / f_fea
    const v2f* f2 = (const v2f*)&feat[(fsel*TILE + pcol) * FSTR];
    const v2f* aw = (const v2f*)&AwLds[(m0 + pcol) * ASTR];

    for (int tl = 0; tl < TILES_PER_WG; ++tl) {
        __syncthreads();   // protect LDS reuse across tiles (also covers initial fills)
        const int tileId = blockIdx.x * TILES_PER_WG + tl;
        const int pix0 = tileId * TILE;
        const int b    = pix0 / HWSZ;
        const int hw0  = pix0 % HWSZ;

        // ---- stage 16 pixels x 256 ch x 3 features, transposed into LDS (b128 loads) ----
        for (int idx = tid; idx < 3*TILE*CH/4; idx += NTHREADS) {
            int jg  = idx & 3;            // 4-pixel group
            int c   = (idx >> 2) & 255;
            int f   = idx >> 10;
            const float* src = (f == 0) ? p_fea : (f == 1) ? h_fea : f_fea;
            float4 v4 = *(const float4*)(src + ((size_t)b*CH + c)*HWSZ + hw0 + jg*4);
            feat[(f*TILE + jg*4 + 0)*FSTR + c] = v4.x;
            feat[(f*TILE + jg*4 + 1)*FSTR + c] = v4.y;
            feat[(f*TILE + jg*4 + 2)*FSTR + c] = v4.z;
            feat[(f*TILE + jg*4 + 3)*FSTR + c] = v4.w;
        }
        __syncthreads();

        // ---- 64 x v_wmma_f32_16x16x4_f32 (K = 256), 4 independent chains ----
        v8f acc0 = {}, acc1 = {}, acc2 = {}, acc3 = {};
#pragma unroll
        for (int s = 0; s < 32; s += 4) {   // K 0..127: A from LDS
            acc0 = __builtin_amdgcn_wmma_f32_16x16x4_f32(false, aw[2*(s+0)+khalf], false, f2[2*(s+0)+khalf], (short)0, acc0, false, false);
            acc1 = __builtin_amdgcn_wmma_f32_16x16x4_f32(false, aw[2*(s+1)+khalf], false, f2[2*(s+1)+khalf], (short)0, acc1, false, false);
            acc2 = __builtin_amdgcn_wmma_f32_16x16x4_f32(false, aw[2*(s+2)+khalf], false, f2[2*(s+2)+khalf], (short)0, acc2, false, false);
            acc3 = __builtin_amdgcn_wmma_f32_16x16x4_f32(false, aw[2*(s+3)+khalf], false, f2[2*(s+3)+khalf], (short)0, acc3, false, false);
        }
#pragma unroll
        for (int s = 32; s < 64; s += 4) {  // K 128..255: A from registers
            acc0 = __builtin_amdgcn_wmma_f32_16x16x4_f32(false, Ar[s-32+0], false, f2[2*(s+0)+khalf], (short)0, acc0, false, false);
            acc1 = __builtin_amdgcn_wmma_f32_16x16x4_f32(false, Ar[s-32+1], false, f2[2*(s+1)+khalf], (short)0, acc1, false, false);
            acc2 = __builtin_amdgcn_wmma_f32_16x16x4_f32(false, Ar[s-32+2], false, f2[2*(s+2)+khalf], (short)0, acc2, false, false);
            acc3 = __builtin_amdgcn_wmma_f32_16x16x4_f32(false, Ar[s-32+3], false, f2[2*(s+3)+khalf], (short)0, acc3, false, false);
        }
        v8f acc = (acc0 + acc1) + (acc2 + acc3);
#pragma unroll
        for (int r = 0; r < 8; ++r)
            Y[(m0 + r + khalf*8)*YSTR + pcol] = acc[r];
        __syncthreads();

        // ---- leaf cells: 6 leaves x HID x 16 pixels ----
        for (int idx = tid; idx < 960; idx += NTHREADS) {
            int p = idx & 15;
            int q = idx >> 4;       // 0..59
            int n = q / 10;
            int j = q % 10;
            float sv = seg_p[((size_t)b*7 + (n+1))*HWSZ + hw0 + p];
            float iv = Y[(n*30 + j     )*YSTR + p] + smallw[n*30 + j     ] * sv;
            float ov = Y[(n*30 + 10 + j)*YSTR + p] + smallw[n*30 + 10 + j] * sv;
            float gv = Y[(n*30 + 20 + j)*YSTR + p] + smallw[n*30 + 20 + j] * sv;
            float c = sigmoidf(iv) * tanhf(gv);
            float h = sigmoidf(ov) * tanhf(c);
            leafH[(n*HID + j)*TILE + p] = h;
            leafC[(n*HID + j)*TILE + p] = c;
            out[(((size_t)(0 + n)*BATCH + b)*HID + j)*HWSZ + hw0 + p] = h;
            out[(((size_t)(9 + n)*BATCH + b)*HID + j)*HWSZ + hw0 + p] = c;
        }
        __syncthreads();

        // ---- upper (children 0..3) and lower (children 4..5) tree cells ----
        if (tid < 320) {
            int p    = tid & 15;
            int q    = tid >> 4;      // 0..19
            int cell = q / 10;        // 0=up, 1=low
            int j    = q % 10;
            int cb   = cell ? 4 : 0;
            int nch  = cell ? 2 : 4;
            int wb   = cell ? 620 : 180;
            int fwb  = cell ? 950 : 510;
            int ib   = 192 + cell*40;
            int fb   = 222 + cell*40;
            float sv = seg_h[((size_t)b*3 + 1 + cell)*HWSZ + hw0 + p];
            float iv = Y[(ib + j     )*YSTR + p] + smallw[wb + (j     )*11] * sv;
            float ov = Y[(ib + 10 + j)*YSTR + p] + smallw[wb + (10 + j)*11] * sv;
            float gv = Y[(ib + 20 + j)*YSTR + p] + smallw[wb + (20 + j)*11] * sv;
            float fx = Y[(fb + j     )*YSTR + p] + smallw[fwb + j*11]       * sv;
#pragma unroll
            for (int t = 0; t < HID; ++t) {
                float hs = 0.0f;
                for (int n = 0; n < nch; ++n) hs += leafH[((cb+n)*HID + t)*TILE + p];
                iv += smallw[wb + (j     )*11 + 1 + t] * hs;
                ov += smallw[wb + (10 + j)*11 + 1 + t] * hs;
                gv += smallw[wb + (20 + j)*11 + 1 + t] * hs;
            }
            float csum = 0.0f;
            for (int n = 0; n < nch; ++n) {
                float fpre = fx;
#pragma unroll
                for (int t = 0; t < HID; ++t)
                    fpre += smallw[fwb + j*11 + 1 + t] * leafH[((cb+n)*HID + t)*TILE + p];
                csum += sigmoidf(fpre) * leafC[((cb+n)*HID + j)*TILE + p];
            }
            float c = sigmoidf(iv) * tanhf(gv) + csum;
            if (cell == 0) {
                upCs[j*TILE + p] = c;
                out[(((size_t)15*BATCH + b)*HID + j)*HWSZ + hw0 + p] = c;
            } else {
                float h = sigmoidf(ov) * tanhf(c);
                lowHs[j*TILE + p] = h;
                lowCs[j*TILE + p] = c;
                out[(((size_t)6 *BATCH + b)*HID + j)*HWSZ + hw0 + p] = h;
                out[(((size_t)7 *BATCH + b)*HID + j)*HWSZ + hw0 + p] = h;
                out[(((size_t)16*BATCH + b)*HID + j)*HWSZ + hw0 + p] = c;
            }
        }
        __syncthreads();

        // ---- full body cell: children h = {low_h, low_h}, c = {up_c, low_c} ----
        if (tid < 160) {
            int p = tid & 15;
            int j = tid >> 4;
            float sv = seg_f[((size_t)b*2 + 1)*HWSZ + hw0 + p];
            float iv = Y[(272 + j)*YSTR + p] + smallw[1060 + (j     )*11] * sv;
            float ov = Y[(282 + j)*YSTR + p] + smallw[1060 + (10 + j)*11] * sv;
            float gv = Y[(292 + j)*YSTR + p] + smallw[1060 + (20 + j)*11] * sv;
            float fx = Y[(302 + j)*YSTR + p] + smallw[1390 + j*11]        * sv;
#pragma unroll
            for (int t = 0; t < HID; ++t) {
                float hl = lowHs[t*TILE + p];
                float hs = 2.0f * hl;        // h_sum over identical children
                iv += smallw[1060 + (j     )*11 + 1 + t] * hs;
                ov += smallw[1060 + (10 + j)*11 + 1 + t] * hs;
                gv += smallw[1060 + (20 + j)*11 + 1 + t] * hs;
                fx += smallw[1390 + j*11 + 1 + t] * hl;  // forget gate sees single child h
            }
            float f = sigmoidf(fx);          // identical for both children
            float c = sigmoidf(iv) * tanhf(gv) + f * (upCs[j*TILE + p] + lowCs[j*TILE + p]);
            float h = sigmoidf(ov) * tanhf(c);
            out[(((size_t)8 *BATCH + b)*HID + j)*HWSZ + hw0 + p] = h;
            out[(((size_t)17*BATCH + b)*HID + j)*HWSZ + hw0 + p] = c;
        }
    }
}

extern "C" void kernel_launch(void* const* d_in, const int* in_sizes, int n_in,
                              void* d_out, int out_size, void* d_ws, size_t ws_size,
                              hipStream_t stream) {
    (void)in_sizes; (void)n_in; (void)d_ws; (void)ws_size; (void)out_size;
    const float* seg_p  = (const float*)d_in[0];
    const float* seg_h  = (const float*)d_in[1];
    const float* seg_f  = (const float*)d_in[2];
    const float* p_fea  = (const float*)d_in[3];
    const float* h_fea  = (const float*)d_in[4];
    const float* f_fea  = (const float*)d_in[5];
    const float* W_leaf = (const float*)d_in[6];
    const float* W_up   = (const float*)d_in[7];
    const float* Wf_up  = (const float*)d_in[8];
    const float* W_low  = (const float*)d_in[9];
    const float* Wf_low = (const float*)d_in[10];
    const float* W_full = (const float*)d_in[11];
    const float* Wf_full= (const float*)d_in[12];
    float* out = (float*)d_out;

    tree_convlstm_fused<<<dim3(NWG), dim3(NTHREADS), 0, stream>>>(
        seg_p, seg_h, seg_f, p_fea, h_fea, f_fea,
        W_leaf, W_up, Wf_up, W_low, Wf_low, W_full, Wf_full, out);
}